// ShortConv_23570780520523
// MI455X (gfx1250) — compile-verified
//
#include <hip/hip_runtime.h>

// ---------------------------------------------------------------------------
// ShortConv fused kernel: LayerNorm(D) -> causal depthwise conv (K=4) -> SiLU
// B=8, T=4096, D=2048, fp32. Bandwidth-bound (~560MB traffic -> ~24us at
// 23.3 TB/s). CDNA5 strategy:
//   * per-wave async global->LDS staging (global_load_async_to_lds_b128)
//     with partial s_wait_asynccnt so LN overlaps in-flight staging
//   * wave-per-row LN (wave32 shuffle reduction), gamma/beta folded into conv
//   * rolling-window depthwise conv, SiLU, non-temporal b128 stores
// ---------------------------------------------------------------------------

namespace sc {
constexpr int B = 8;
constexpr int T = 4096;
constexpr int D = 2048;
constexpr int TILE_T = 16;
constexpr int HALO = 3;              // K-1
constexpr int ROWS = TILE_T + HALO;  // 19 rows * 8KB = 152KB LDS
constexpr float EPS = 1e-5f;
}  // namespace sc

#if defined(__gfx1250__) && __has_builtin(__builtin_amdgcn_global_load_async_to_lds_b128)
#define USE_ASYNC_LDS 1
#else
#define USE_ASYNC_LDS 0
#endif

typedef float v4f __attribute__((ext_vector_type(4)));
typedef int v4i __attribute__((ext_vector_type(4)));
typedef __attribute__((address_space(1))) v4i gv4i_t;  // global (AS1)
typedef __attribute__((address_space(3))) v4i lv4i_t;  // LDS    (AS3)

__device__ __forceinline__ void stage16(const float* __restrict__ g, float* l) {
#if USE_ASYNC_LDS
  // per-lane 16B async copy global->LDS; tracked by ASYNCcnt
  __builtin_amdgcn_global_load_async_to_lds_b128((gv4i_t*)g, (lv4i_t*)l,
                                                 /*offset=*/0, /*cpol=*/0);
#else
  *(v4f*)l = *(const v4f*)g;
#endif
}

template <int N>
__device__ __forceinline__ void wait_async_le() {
#if USE_ASYNC_LDS
#if __has_builtin(__builtin_amdgcn_s_wait_asynccnt)
  __builtin_amdgcn_s_wait_asynccnt(N);
#else
  asm volatile("s_wait_asynccnt %0" ::"i"(N) : "memory");
#endif
  asm volatile("" ::: "memory");  // keep LDS reads below the wait
#endif
}

__global__ __launch_bounds__(256) void ShortConv_23570780520523_kernel(
    const float* __restrict__ x, const float* __restrict__ w,
    const float* __restrict__ gamma, const float* __restrict__ beta,
    float* __restrict__ out) {
  using namespace sc;
  __shared__ float sm[ROWS * D];  // raw x tile -> (x-mu)*rstd in place

  const int tid = threadIdx.x;
  const int lane = tid & 31;
  const int wav = tid >> 5;
  const int tiles_per_b = T / TILE_T;  // 256
  const int b = blockIdx.x / tiles_per_b;
  const int t0 = (blockIdx.x % tiles_per_b) * TILE_T;

  // ------------------------------------------------------------ Phase 0 + 1
  // Wave w owns rows {w, w+8, w+16<19}. It async-stages them (16 x b128 per
  // lane per row, chunk pattern lane+32j == the LN read pattern), then runs
  // LN per row gated by partial ASYNCcnt waits (loads complete in issue
  // order; zero-pad rows issue no asyncs and are always first, so the
  // constant thresholds remain conservative-correct).
  const int nrows = (wav + 16 < ROWS) ? 3 : 2;

  for (int k = 0; k < nrows; ++k) {
    const int r = wav + 8 * k;
    const int tr = t0 - HALO + r;
    float* lrow = &sm[r * D];
    if (tr >= 0) {
      const float* grow = x + ((size_t)b * T + tr) * (size_t)D;
#pragma unroll
      for (int j = 0; j < 16; ++j) {
        const int off = (lane + 32 * j) * 4;
        stage16(grow + off, lrow + off);
      }
    } else {  // reference zero-pads AFTER LayerNorm -> rows stay zero
      const v4f z = {0.f, 0.f, 0.f, 0.f};
#pragma unroll
      for (int j = 0; j < 16; ++j) *(v4f*)(lrow + (lane + 32 * j) * 4) = z;
    }
  }

  // LN one row per wait step; store (x-mean)*rstd (gamma/beta folded later)
  auto ln_row = [&](int r) {
    const int tr = t0 - HALO + r;
    if (tr < 0) return;
    float* lrow = &sm[r * D];
    v4f v[16];
    float s = 0.f, ss = 0.f;
#pragma unroll
    for (int j = 0; j < 16; ++j) {
      v[j] = *(const v4f*)(lrow + (lane + j * 32) * 4);
      s += v[j].x + v[j].y + v[j].z + v[j].w;
      ss += v[j].x * v[j].x + v[j].y * v[j].y + v[j].z * v[j].z + v[j].w * v[j].w;
    }
#pragma unroll
    for (int o = 16; o > 0; o >>= 1) {
      s += __shfl_xor(s, o, 32);
      ss += __shfl_xor(ss, o, 32);
    }
    const float mean = s * (1.0f / (float)D);
    const float var = ss * (1.0f / (float)D) - mean * mean;
    const float rstd = rsqrtf(var + EPS);
#pragma unroll
    for (int j = 0; j < 16; ++j) {
      v4f o4;
      o4.x = (v[j].x - mean) * rstd;
      o4.y = (v[j].y - mean) * rstd;
      o4.z = (v[j].z - mean) * rstd;
      o4.w = (v[j].w - mean) * rstd;
      *(v4f*)(lrow + (lane + j * 32) * 4) = o4;
    }
  };

  if (nrows == 3) {
    wait_async_le<32>();
    ln_row(wav);
    wait_async_le<16>();
    ln_row(wav + 8);
    wait_async_le<0>();
    ln_row(wav + 16);
  } else {
    wait_async_le<16>();
    ln_row(wav);
    wait_async_le<0>();
    ln_row(wav + 8);
  }
  __syncthreads();

  // ------------------------------------------------------------------ Phase 2
  // Depthwise conv + folded gamma/beta + SiLU. Thread owns 2 chunks of 4
  // channels. conv_w is (D,4) row-major -> taps of channel d are one aligned
  // float4. y = gamma * sum_j(w_j * xhat) + beta * sum_{valid j}(w_j).
#pragma unroll
  for (int i = 0; i < 2; ++i) {
    const int d0 = (tid + i * 256) * 4;
    const v4f w0 = *(const v4f*)(w + (size_t)(d0 + 0) * 4);  // taps of ch d0
    const v4f w1 = *(const v4f*)(w + (size_t)(d0 + 1) * 4);
    const v4f w2 = *(const v4f*)(w + (size_t)(d0 + 2) * 4);
    const v4f w3 = *(const v4f*)(w + (size_t)(d0 + 3) * 4);
    const v4f g4 = *(const v4f*)(gamma + d0);
    const v4f b4 = *(const v4f*)(beta + d0);

    // valid-tap weight sums per channel (for the t<3 zero-padded prefix)
    v4f swA = {w0.w, w1.w, w2.w, w3.w};  // t==0 : tap 3 only
    v4f tz = {w0.z, w1.z, w2.z, w3.z};
    v4f ty = {w0.y, w1.y, w2.y, w3.y};
    v4f tx = {w0.x, w1.x, w2.x, w3.x};
    v4f swB = swA + tz;        // t==1
    v4f swC = swB + ty;        // t==2
    v4f swD = swC + tx;        // t>=3 : all taps
    const v4f btA = b4 * swA, btB = b4 * swB, btC = b4 * swC, btD = b4 * swD;

    v4f a = *(const v4f*)(&sm[0 * D + d0]);
    v4f bq = *(const v4f*)(&sm[1 * D + d0]);
    v4f cq = *(const v4f*)(&sm[2 * D + d0]);
#pragma unroll
    for (int ro = 0; ro < TILE_T; ++ro) {
      const v4f dq = *(const v4f*)(&sm[(ro + 3) * D + d0]);
      v4f acc;
      acc.x = w0.x * a.x + w0.y * bq.x + w0.z * cq.x + w0.w * dq.x;
      acc.y = w1.x * a.y + w1.y * bq.y + w1.z * cq.y + w1.w * dq.y;
      acc.z = w2.x * a.z + w2.y * bq.z + w2.z * cq.z + w2.w * dq.z;
      acc.w = w3.x * a.w + w3.y * bq.w + w3.z * cq.w + w3.w * dq.w;

      v4f bt = btD;
      if (t0 == 0 && ro < 3) {  // only tile 0: fewer valid taps for beta term
        bt = (ro == 0) ? btA : (ro == 1) ? btB : btC;
      }
      v4f y = g4 * acc + bt;

      // SiLU: y * sigmoid(y)
      y.x *= 1.0f / (1.0f + __expf(-y.x));
      y.y *= 1.0f / (1.0f + __expf(-y.y));
      y.z *= 1.0f / (1.0f + __expf(-y.z));
      y.w *= 1.0f / (1.0f + __expf(-y.w));

      // write-once output: non-temporal so it doesn't evict halo lines in L2
      __builtin_nontemporal_store(
          y, (v4f*)(out + ((size_t)b * T + (size_t)(t0 + ro)) * (size_t)D + d0));
      a = bq;
      bq = cq;
      cq = dq;
    }
  }
}

extern "C" void kernel_launch(void* const* d_in, const int* in_sizes, int n_in,
                              void* d_out, int out_size, void* d_ws, size_t ws_size,
                              hipStream_t stream) {
  (void)in_sizes;
  (void)n_in;
  (void)out_size;
  (void)d_ws;
  (void)ws_size;
  const float* x = (const float*)d_in[0];
  const float* w = (const float*)d_in[1];      // (D, K) row-major
  const float* gamma = (const float*)d_in[2];  // (D,)
  const float* beta = (const float*)d_in[3];   // (D,)
  float* out = (float*)d_out;

  const int grid = sc::B * (sc::T / sc::TILE_T);  // 2048 blocks
  ShortConv_23570780520523_kernel<<<grid, 256, 0, stream>>>(x, w, gamma, beta, out);
}